// EdgeCentral_14405320311140
// MI455X (gfx1250) — compile-verified
//
#include <hip/hip_runtime.h>
#include <hip/hip_bf16.h>

typedef __attribute__((ext_vector_type(2))) float v2f;
typedef __attribute__((ext_vector_type(4))) float v4f;
typedef __attribute__((ext_vector_type(8))) float v8f;

#define DD    64
#define NCOL  4096   // D*D
#define KDIM  128    // concat(atom_feat, bond_feat)
#define AP    132    // padded LDS row stride (floats) -> conflict-free
#define MT    64     // edges per block (4 WMMA M-tiles)

// ---------------------------------------------------------------------------
// K12[k, h*64+d] = sum_f W[h, f + (k>=64 ? 64 : 0)] * kernel[k&63, f*64+d]
// Folds weight_edge into kernel: removes the need to materialize
// atoms_hid / bond_hid (245 MB) entirely.
// ---------------------------------------------------------------------------
__global__ void __launch_bounds__(256)
prep_k12(const float* __restrict__ kern, const float* __restrict__ W,
         float* __restrict__ K12) {
    int n = blockIdx.x * blockDim.x + threadIdx.x;   // 0..4095
    int k = blockIdx.y;                              // 0..127
    int h = n >> 6, d = n & 63;
    const float* wrow = W + h * 128 + ((k >= 64) ? 64 : 0);
    const float* krow = kern + (size_t)(k & 63) * NCOL + d;
    float acc = 0.f;
#pragma unroll 8
    for (int f = 0; f < 64; ++f) acc += wrow[f] * krow[f * 64];
    K12[(size_t)k * NCOL + n] = acc;
}

// Bc[h*64+d] = sum_f (W[h,f] + W[h,64+f]) * bias[f*64+d]
__global__ void __launch_bounds__(256)
prep_bc(const float* __restrict__ bias, const float* __restrict__ W,
        float* __restrict__ Bc) {
    int n = blockIdx.x * blockDim.x + threadIdx.x;   // 0..4095
    int h = n >> 6, d = n & 63;
    float acc = 0.f;
#pragma unroll 8
    for (int f = 0; f < 64; ++f)
        acc += (W[h * 128 + f] + W[h * 128 + 64 + f]) * bias[f * 64 + d];
    Bc[n] = acc;
}

// ev[e,h] = relu(sum_d Winp[h,d] * bonds[pairs[e,1], d])
__global__ void __launch_bounds__(256)
edge_vec(const float* __restrict__ bonds, const int* __restrict__ pairs,
         const float* __restrict__ Winp, float* __restrict__ ev, int E) {
    int t = blockIdx.x * blockDim.x + threadIdx.x;
    if (t >= E * 64) return;
    int e = t >> 6, h = t & 63;
    const float* br = bonds + (size_t)pairs[e * 2 + 1] * DD;
    const float* wr = Winp + h * DD;
    float acc = 0.f;
#pragma unroll 8
    for (int d = 0; d < DD; ++d) acc += wr[d] * br[d];
    ev[t] = fmaxf(acc, 0.f);
}

// ---------------------------------------------------------------------------
// Main GEMM:  result[e, n] = relu( x_e(1x128) @ K12(128x4096) + Bc[n] + ev[e,h] )
// x_e = [atoms[pairs[e,0]] | bonds[e]]   ; written 3x (STEPS stack).
// Block = 64 edges x 128 cols ; 8 waves. Each wave owns one 16-col chunk and
// computes FOUR 16x16 f32 WMMA tiles (row groups m0+0/16/32/48) sharing the
// same B-fragment stream: 4 wmma per 2 B-loads (2:1), ~320 MB total L2 B
// traffic, and four independent accumulator chains interleave on the XDL pipe.
// ---------------------------------------------------------------------------
__global__ void __launch_bounds__(256)
edge_gemm(const float* __restrict__ atoms, const float* __restrict__ bonds,
          const int* __restrict__ pairs, const float* __restrict__ K12,
          const float* __restrict__ Bc, const float* __restrict__ ev,
          float* __restrict__ out, int E) {
    __shared__ float Atile[MT * AP];     // [m][k], padded (33.8 KB)

    const int m0  = blockIdx.y * MT;     // edge tile base
    const int n0  = blockIdx.x * 128;    // column tile base
    const int tid = threadIdx.x;

    // ---- stage A tile: 64 edges x 128 feats = 8192 floats, 32 per thread ----
    {
        int m = tid >> 2;                // 0..63 row (4 threads per row)
        int c = (tid & 3) * 32;          // 0,32,64,96 (chunk within one src)
        int e = m0 + m; if (e >= E) e = E - 1;
        const float* src = (c < 64)
            ? (atoms + (size_t)pairs[e * 2] * DD + c)
            : (bonds + (size_t)e * DD + (c - 64));
#pragma unroll
        for (int j = 0; j < 8; ++j)
            *(v4f*)&Atile[m * AP + c + 4 * j] = *(const v4f*)(src + 4 * j);
    }
    __syncthreads();

    const int wave = tid >> 5;           // 0..7
    const int lane = tid & 31;
    const int half = lane >> 4;          // 0 | 1
    const int l    = lane & 15;
    const int colBase = n0 + wave * 16;
    const float* Bp = K12 + colBase + l; // B[k][col] = K12[k*4096 + col]

    v8f acc0 = {};                       // rows m0      .. m0+15
    v8f acc1 = {};                       // rows m0+16   .. m0+31
    v8f acc2 = {};                       // rows m0+32   .. m0+47
    v8f acc3 = {};                       // rows m0+48   .. m0+63
#pragma unroll
    for (int k = 0; k < KDIM; k += 4) {
        int ka = k + half * 2;           // ISA 16x16x4 f32 lane layout
        v2f a0, a1, a2, a3, b;
        b.x  = Bp[(size_t)ka * NCOL];
        b.y  = Bp[(size_t)(ka + 1) * NCOL];
        a0.x = Atile[l * AP + ka];        a0.y = Atile[l * AP + ka + 1];
        a1.x = Atile[(16 + l) * AP + ka]; a1.y = Atile[(16 + l) * AP + ka + 1];
        a2.x = Atile[(32 + l) * AP + ka]; a2.y = Atile[(32 + l) * AP + ka + 1];
        a3.x = Atile[(48 + l) * AP + ka]; a3.y = Atile[(48 + l) * AP + ka + 1];
        acc0 = __builtin_amdgcn_wmma_f32_16x16x4_f32(
            false, a0, false, b, (short)0, acc0, false, false);
        acc1 = __builtin_amdgcn_wmma_f32_16x16x4_f32(
            false, a1, false, b, (short)0, acc1, false, false);
        acc2 = __builtin_amdgcn_wmma_f32_16x16x4_f32(
            false, a2, false, b, (short)0, acc2, false, false);
        acc3 = __builtin_amdgcn_wmma_f32_16x16x4_f32(
            false, a3, false, b, (short)0, acc3, false, false);
    }

    // ---- epilogue: + Bc + ev, relu, 3 stacked NT stores ----
    const int col = colBase + l;
    const int h   = colBase >> 6;        // constant within a 16-col wave tile
    const float bc = Bc[col];
    const size_t plane = (size_t)E * NCOL;
#pragma unroll
    for (int t = 0; t < 4; ++t) {
        v8f acc = (t == 0) ? acc0 : (t == 1) ? acc1 : (t == 2) ? acc2 : acc3;
#pragma unroll
        for (int v = 0; v < 8; ++v) {
            int e = m0 + t * 16 + v + half * 8;  // D-matrix row mapping
            if (e < E) {
                float r = fmaxf(acc[v] + bc + ev[(size_t)e * 64 + h], 0.f);
                size_t off = (size_t)e * NCOL + col;
                __builtin_nontemporal_store(r, &out[off]);
                __builtin_nontemporal_store(r, &out[off + plane]);
                __builtin_nontemporal_store(r, &out[off + 2 * plane]);
            }
        }
    }
}

// ---------------------------------------------------------------------------
extern "C" void kernel_launch(void* const* d_in, const int* in_sizes, int n_in,
                              void* d_out, int out_size, void* d_ws, size_t ws_size,
                              hipStream_t stream) {
    const float* atoms = (const float*)d_in[0];   // (N, 64)
    const float* bonds = (const float*)d_in[1];   // (E, 64)
    const int*   pairs = (const int*)  d_in[2];   // (E, 2)
    const float* kern  = (const float*)d_in[3];   // (64, 4096)
    const float* bias  = (const float*)d_in[4];   // (4096,)
    const float* Wedge = (const float*)d_in[5];   // (64, 128)
    const float* Winp  = (const float*)d_in[6];   // (64, 64)
    float* out = (float*)d_out;

    const int E = in_sizes[1] / DD;

    // workspace: K12 (128*4096 f) | Bc (4096 f) | ev (E*64 f)  ~= 4.7 MB
    float* K12 = (float*)d_ws;
    float* Bc  = K12 + (size_t)KDIM * NCOL;
    float* ev  = Bc + NCOL;

    prep_k12<<<dim3(NCOL / 256, KDIM), 256, 0, stream>>>(kern, Wedge, K12);
    prep_bc <<<dim3(NCOL / 256),       256, 0, stream>>>(bias, Wedge, Bc);
    edge_vec<<<dim3((E * 64 + 255) / 256), 256, 0, stream>>>(bonds, pairs, Winp, ev, E);
    edge_gemm<<<dim3(NCOL / 128, (E + MT - 1) / MT), 256, 0, stream>>>(
        atoms, bonds, pairs, K12, Bc, ev, out, E);
}